// Dynamic_deformablev2_40338332844081
// MI455X (gfx1250) — compile-verified
//
#include <hip/hip_runtime.h>

#define Hc 480
#define Wc 640
#define Bc 4
#define NPIX (Bc * Hc * Wc)
#define TILE 128

typedef float v2f __attribute__((ext_vector_type(2)));
typedef float v8f __attribute__((ext_vector_type(8)));

// ---------------------------------------------------------------------------
// Stage 1: offset/affinity convs (8ch -> 24ch, 3x3, K=72) via f32 WMMA.
// One block = 1 row-segment of 128 pixels; 8 waves, each wave does a
// 16-pixel x 24-outch tile with V_WMMA_F32_16X16X4_F32 (exact f32 MACs).
// ---------------------------------------------------------------------------
__global__ __launch_bounds__(256) void offset_conv_wmma(
    const float* __restrict__ guidance,
    const float* __restrict__ w1, const float* __restrict__ b1,
    const float* __restrict__ w2, const float* __restrict__ b2,
    float* __restrict__ oa1, float* __restrict__ oa2)
{
    __shared__ float sG1[8][3][TILE + 4];
    __shared__ float sG2[8][3][TILE + 4];
    __shared__ float sW1[72][32];
    __shared__ float sW2[72][32];

    const int tid = threadIdx.x;
    const int xb  = blockIdx.x * TILE;
    const int y   = blockIdx.y;
    const int b   = blockIdx.z;

    // Weights -> LDS as [K=72][N padded to 32].  w layout (24,8,3,3).
    for (int idx = tid; idx < 72 * 32; idx += 256) {
        int k = idx >> 5, n = idx & 31;
        float v1 = 0.f, v2 = 0.f;
        if (n < 24) {
            int i = k / 9, pos = k % 9;
            int widx = (n * 8 + i) * 9 + pos;
            v1 = w1[widx];
            v2 = w2[widx];
        }
        sW1[k][n] = v1;
        sW2[k][n] = v2;
    }
    // Guidance halo tile: g1 = ch0..7, g2 = ch8..15; rows y-1..y+1, cols xb-1..xb+128
    for (int idx = tid; idx < 8 * 3 * (TILE + 2); idx += 256) {
        int c  = idx % (TILE + 2);
        int t  = idx / (TILE + 2);
        int r  = t % 3;
        int ch = t / 3;
        int yy = y + r - 1;
        int xx = xb + c - 1;
        float v1 = 0.f, v2 = 0.f;
        if (yy >= 0 && yy < Hc && xx >= 0 && xx < Wc) {
            size_t base = (((size_t)b * 24 + ch) * Hc + yy) * Wc + xx;
            v1 = guidance[base];
            v2 = guidance[base + (size_t)8 * Hc * Wc];
        }
        sG1[ch][r][c] = v1;
        sG2[ch][r][c] = v2;
    }
    __syncthreads();

    const int wave = tid >> 5;
    const int lane = tid & 31;
    const int hh   = lane >> 4;   // lane half
    const int ln   = lane & 15;
    const int lx   = wave * 16;   // this wave's 16-pixel tile base (local)

    v8f c1a = {}, c1b = {}, c2a = {}, c2b = {};
#pragma unroll
    for (int k0 = 0; k0 < 72; k0 += 4) {
        int ka = k0 + 2 * hh;
        int kb = ka + 1;
        int ia = ka / 9, pa = ka % 9, ya_ = pa / 3, xa_ = pa % 3;
        int ib = kb / 9, pb = kb % 9, yb_ = pb / 3, xb_ = pb % 3;
        v2f A1, A2, B10, B11, B20, B21;
        // A: M = ln (pixel), K = ka/kb.  patch(M,K) = g[i][ky][localx+kx]
        A1.x = sG1[ia][ya_][lx + ln + xa_];
        A1.y = sG1[ib][yb_][lx + ln + xb_];
        A2.x = sG2[ia][ya_][lx + ln + xa_];
        A2.y = sG2[ib][yb_][lx + ln + xb_];
        // B: N = ln (+16 for second tile), K = ka/kb
        B10.x = sW1[ka][ln];        B10.y = sW1[kb][ln];
        B11.x = sW1[ka][16 + ln];   B11.y = sW1[kb][16 + ln];
        B20.x = sW2[ka][ln];        B20.y = sW2[kb][ln];
        B21.x = sW2[ka][16 + ln];   B21.y = sW2[kb][16 + ln];
        c1a = __builtin_amdgcn_wmma_f32_16x16x4_f32(false, A1, false, B10, (short)0, c1a, false, false);
        c1b = __builtin_amdgcn_wmma_f32_16x16x4_f32(false, A1, false, B11, (short)0, c1b, false, false);
        c2a = __builtin_amdgcn_wmma_f32_16x16x4_f32(false, A2, false, B20, (short)0, c2a, false, false);
        c2b = __builtin_amdgcn_wmma_f32_16x16x4_f32(false, A2, false, B21, (short)0, c2b, false, false);
    }

    // D layout: M = v + 8*hh, N = ln (tile0) / 16+ln (tile1)
    float bias1_0 = b1[ln];
    float bias2_0 = b2[ln];
    float bias1_1 = (ln < 8) ? b1[16 + ln] : 0.f;
    float bias2_1 = (ln < 8) ? b2[16 + ln] : 0.f;
#pragma unroll
    for (int v = 0; v < 8; ++v) {
        int p = lx + v + 8 * hh;
        size_t pix = ((size_t)b * Hc + y) * Wc + xb + p;
        oa1[pix * 24 + ln] = c1a[v] + bias1_0;
        oa2[pix * 24 + ln] = c2a[v] + bias2_0;
        if (ln < 8) {
            oa1[pix * 24 + 16 + ln] = c1b[v] + bias1_1;
            oa2[pix * 24 + 16 + ln] = c2b[v] + bias2_1;
        }
    }
}

// ---------------------------------------------------------------------------
// Stage 1b: fold all loop-invariants into per-pixel coefficients.
// For each k: c0..c3 = att/gsa, r = (gsa - att.aff_sum)*feat_init/gsa.
// ---------------------------------------------------------------------------
__global__ __launch_bounds__(256) void precompute_kernel(
    const float* __restrict__ feat_init, const float* __restrict__ guidance,
    const float* __restrict__ dynamic, const float* __restrict__ confidence,
    const float* __restrict__ feat_fix,
    const float* __restrict__ oa1, const float* __restrict__ oa2,
    float* __restrict__ coef, float* __restrict__ mc_arr,
    float* __restrict__ mcf_arr, float* __restrict__ featA)
{
    int pix = blockIdx.x * 256 + threadIdx.x;
    if (pix >= NPIX) return;
    int x = pix % Wc;
    int y = (pix / Wc) % Hc;
    int b = pix / (Wc * Hc);

    float s1 = 0.f, a1 = 0.f, s2 = 0.f, a2 = 0.f;
    const float* p1 = oa1 + (size_t)pix * 24 + 16;
    const float* p2 = oa2 + (size_t)pix * 24 + 16;
#pragma unroll
    for (int i = 0; i < 8; ++i) {
        float v = p1[i]; s1 += v; a1 += fabsf(v);
        v = p2[i];       s2 += v; a2 += fabsf(v);
    }

    // sumw "conv": 8-neighbor shifted sum of g3 (channel i -> offset (1-j/3, 1-j%3))
    float gsum = 0.f, gabs = 0.f;
#pragma unroll
    for (int i = 0; i < 8; ++i) {
        int j  = (i < 4) ? i : i + 1;
        int dy = 1 - j / 3, dx = 1 - j % 3;
        int yy = y + dy, xx = x + dx;
        if (yy >= 0 && yy < Hc && xx >= 0 && xx < Wc) {
            float g = guidance[(((size_t)b * 24 + 16 + i) * Hc + yy) * Wc + xx];
            gsum += g;
            gabs += fabsf(g);
        }
    }

    float conf = 1.f / (1.f + expf(-confidence[pix]));
    float ff   = feat_fix[pix];
    float sgn  = (ff > 0.f) ? 1.f : ((ff < 0.f) ? -1.f : 0.f);
    float mc   = sgn * conf;
    float fi   = feat_init[pix];
    featA[pix]   = fi;
    mc_arr[pix]  = mc;
    mcf_arr[pix] = mc * ff;

#pragma unroll
    for (int k = 0; k < 6; ++k) {
        float d0 = dynamic[(((size_t)b * 24 + 4 * k + 0) * Hc + y) * Wc + x];
        float d1 = dynamic[(((size_t)b * 24 + 4 * k + 1) * Hc + y) * Wc + x];
        float d2 = dynamic[(((size_t)b * 24 + 4 * k + 2) * Hc + y) * Wc + x];
        float d3 = dynamic[(((size_t)b * 24 + 4 * k + 3) * Hc + y) * Wc + x];
        float mx = fmaxf(fmaxf(d0, d1), fmaxf(d2, d3));
        float e0 = expf(d0 - mx), e1 = expf(d1 - mx);
        float e2 = expf(d2 - mx), e3 = expf(d3 - mx);
        float ie = 1.f / (e0 + e1 + e2 + e3);
        float t0 = e0 * ie, t1 = e1 * ie, t2 = e2 * ie, t3 = e3 * ie;
        float gsa  = t0 * a1 + t1 * a2 + t2 * gabs + t3 + 1e-4f;
        float asum = t0 * s1 + t1 * s2 + t2 * gsum + t3;
        float ig = 1.f / gsa;
        coef[(size_t)(k * 5 + 0) * NPIX + pix] = t0 * ig;
        coef[(size_t)(k * 5 + 1) * NPIX + pix] = t1 * ig;
        coef[(size_t)(k * 5 + 2) * NPIX + pix] = t2 * ig;
        coef[(size_t)(k * 5 + 3) * NPIX + pix] = t3 * ig;
        coef[(size_t)(k * 5 + 4) * NPIX + pix] = (gsa - asum) * ig * fi;
    }
}

// ---------------------------------------------------------------------------
// Stage 2: one propagation iteration (run 6x, ping-pong).
// Center tap of the deformable conv has affinity 0 (inserted ref) -> skipped.
// ---------------------------------------------------------------------------
__device__ __forceinline__ float samp(const float* __restrict__ f, int yi, int xi)
{
    if (yi < 0 || yi >= Hc || xi < 0 || xi >= Wc) return 0.f;
    return f[yi * Wc + xi];
}

__device__ __forceinline__ float deform8(const float* __restrict__ f,
                                         const float* __restrict__ oa,
                                         int y, int x)
{
    float acc = 0.f;
#pragma unroll
    for (int t = 0; t < 8; ++t) {
        int k9 = (t < 4) ? t : t + 1;
        int ky = k9 / 3, kx = k9 % 3;
        float py = (float)y + (float)(ky - 1) + oa[2 * t];
        float px = (float)x + (float)(kx - 1) + oa[2 * t + 1];
        float y0f = floorf(py), x0f = floorf(px);
        int   y0  = (int)y0f,  x0  = (int)x0f;
        float wy = py - y0f, wx = px - x0f;
        float s = (1.f - wy) * (1.f - wx) * samp(f, y0,     x0)
                + (1.f - wy) * wx         * samp(f, y0,     x0 + 1)
                + wy * (1.f - wx)         * samp(f, y0 + 1, x0)
                + wy * wx                 * samp(f, y0 + 1, x0 + 1);
        acc += oa[16 + t] * s;
    }
    return acc;
}

__global__ __launch_bounds__(256) void prop_step(
    const float* __restrict__ src, float* __restrict__ dst,
    const float* __restrict__ guidance,
    const float* __restrict__ oa1, const float* __restrict__ oa2,
    const float* __restrict__ coef, const float* __restrict__ mc_arr,
    const float* __restrict__ mcf_arr, int k)
{
    int pix = blockIdx.x * 256 + threadIdx.x;
    if (pix >= NPIX) return;
    int x = pix % Wc;
    int y = (pix / Wc) % Hc;
    int b = pix / (Wc * Hc);

    const float* fplane = src + (size_t)b * Hc * Wc;

    float acc1 = deform8(fplane, oa1 + (size_t)pix * 24, y, x);
    float acc2 = deform8(fplane, oa2 + (size_t)pix * 24, y, x);

    float acc3 = 0.f;
#pragma unroll
    for (int i = 0; i < 8; ++i) {
        int j  = (i < 4) ? i : i + 1;
        int dy = 1 - j / 3, dx = 1 - j % 3;
        int yy = y + dy, xx = x + dx;
        if (yy >= 0 && yy < Hc && xx >= 0 && xx < Wc) {
            float g = guidance[(((size_t)b * 24 + 16 + i) * Hc + yy) * Wc + xx];
            acc3 += g * fplane[yy * Wc + xx];
        }
    }

    float c0 = coef[(size_t)(k * 5 + 0) * NPIX + pix];
    float c1 = coef[(size_t)(k * 5 + 1) * NPIX + pix];
    float c2 = coef[(size_t)(k * 5 + 2) * NPIX + pix];
    float c3 = coef[(size_t)(k * 5 + 3) * NPIX + pix];
    float r  = coef[(size_t)(k * 5 + 4) * NPIX + pix];

    float fv = src[pix];
    float fn = c0 * acc1 + c1 * acc2 + c2 * acc3 + c3 * fv + r;
    float mc = mc_arr[pix];
    dst[pix] = (1.f - mc) * fn + mcf_arr[pix];
}

// ---------------------------------------------------------------------------
extern "C" void kernel_launch(void* const* d_in, const int* in_sizes, int n_in,
                              void* d_out, int out_size, void* d_ws, size_t ws_size,
                              hipStream_t stream)
{
    const float* feat_init  = (const float*)d_in[0];
    const float* guidance   = (const float*)d_in[1];
    const float* dynamic    = (const float*)d_in[2];
    const float* confidence = (const float*)d_in[3];
    const float* feat_fix   = (const float*)d_in[4];
    const float* w1 = (const float*)d_in[5];
    const float* b1 = (const float*)d_in[6];
    const float* w2 = (const float*)d_in[7];
    const float* b2 = (const float*)d_in[8];
    float* out = (float*)d_out;

    float* ws   = (float*)d_ws;
    float* OA1  = ws;                              // NPIX*24
    float* OA2  = OA1 + (size_t)NPIX * 24;         // NPIX*24
    float* COEF = OA2 + (size_t)NPIX * 24;         // NPIX*30 (6 k x 5, planar)
    float* MC   = COEF + (size_t)NPIX * 30;        // NPIX
    float* MCF  = MC + (size_t)NPIX;               // NPIX
    float* FA   = MCF + (size_t)NPIX;              // NPIX
    float* FB   = FA + (size_t)NPIX;               // NPIX

    dim3 gconv(Wc / TILE, Hc, Bc);
    offset_conv_wmma<<<gconv, 256, 0, stream>>>(guidance, w1, b1, w2, b2, OA1, OA2);

    int nb = (NPIX + 255) / 256;
    precompute_kernel<<<nb, 256, 0, stream>>>(feat_init, guidance, dynamic, confidence,
                                              feat_fix, OA1, OA2, COEF, MC, MCF, FA);

    const float* src = FA;
    for (int k = 0; k < 6; ++k) {
        float* dst = (k == 5) ? out : ((k % 2 == 0) ? FB : FA);
        prop_step<<<nb, 256, 0, stream>>>(src, dst, guidance, OA1, OA2, COEF, MC, MCF, k);
        src = dst;
    }
}